// SSIMLoss_55929064128588
// MI455X (gfx1250) — compile-verified
//
#include <hip/hip_runtime.h>
#include <hip/hip_bf16.h>

typedef __attribute__((ext_vector_type(2))) float v2f;
typedef __attribute__((ext_vector_type(8))) float v8f;

// Normalized 11-tap Gaussian, sigma = 1.5 (matches reference _gaussian_kernel()).
__device__ __constant__ float GAUSS11[11] = {
    0.00102838f, 0.00759876f, 0.03600077f, 0.10936073f, 0.21300547f,
    0.26601172f,
    0.21300547f, 0.10936073f, 0.03600077f, 0.00759876f, 0.00102838f};

#define WMMA_F32X4(A, B, C)                                                   \
    __builtin_amdgcn_wmma_f32_16x16x4_f32(false, (A), false, (B), (short)0,   \
                                          (C), false, false)

__device__ __forceinline__ float wgt(const float* G, int idx) {
    return ((unsigned)idx <= 10u) ? G[idx] : 0.0f;
}

// One wave32 computes one 16x16 output tile of one (b,c) image plane.
// Separable 11-tap blur as banded-matrix GEMMs on the WMMA pipe:
//   H_f[32x16]  = T_f[32x28] x Wh[28x16]   (2 row-tiles x 7 K-chunks)
//   Out_f[16x16]= Wv[16x28]  x H_f[28x16]  (7 K-chunks)
// All 5 fields {x, y, x*x, y*y, x*y} share one set of A-fragment LDS loads
// in the horizontal pass (5 live accumulators per row-tile).
__global__ __launch_bounds__(64) void ssim_wmma_kernel(
    const float* __restrict__ den, const float* __restrict__ cln,
    float* __restrict__ accum, int nTiles) {
    // Per-wave LDS (floats): Tx 32x28=896, Ty 896, Ht 5 x (16 cols x 34)=2720
    __shared__ float smem[2 * 4512 + 16];
    const int lane = threadIdx.x & 31;
    const int wave = threadIdx.x >> 5;
    float* Tx = smem + wave * 4512;
    float* Ty = Tx + 896;
    float* Ht = Tx + 1792;              // transposed H: Ht[f*544 + col*34 + row]
    float* G  = smem + 2 * 4512;        // gauss taps, shared by block

    if (threadIdx.x < 16)
        G[threadIdx.x] = (threadIdx.x < 11) ? GAUSS11[threadIdx.x] : 0.0f;
    __syncthreads();

    const int tile = blockIdx.x * 2 + wave;
    if (tile >= nTiles) return;

    const int tx = tile & 31;
    const int ty = (tile >> 5) & 31;
    const int plane = tile >> 10;       // b*C + c, 0..95
    const float* __restrict__ pd = den + (size_t)plane * (512 * 512);
    const float* __restrict__ pc = cln + (size_t)plane * (512 * 512);
    const int row0 = ty * 16 - 5;       // halo = WINDOW/2 = 5
    const int col0 = tx * 16 - 5;

    // ---- Stage clipped inputs. Only the 26x26 halo region contributes
    // (Gaussian band is zero for r,c >= 26); the rest is zero-filled WITHOUT
    // a global load, but must be written (NaN garbage would poison WMMA).
    for (int i = lane; i < 32 * 28; i += 32) {
        const int r = i / 28;
        const int c = i - r * 28;
        float vx = 0.0f, vy = 0.0f;
        if (r < 26 && c < 26) {
            const int gr = row0 + r;
            const int gc = col0 + c;
            if ((unsigned)gr < 512u && (unsigned)gc < 512u) {
                const size_t off = (size_t)gr * 512 + gc;
                vx = fminf(fmaxf(pd[off], 0.0f), 1.0f);
                vy = fminf(fmaxf(pc[off], 0.0f), 1.0f);
            }
        }
        Tx[i] = vx;
        Ty[i] = vy;
    }

    const int half = lane >> 4;         // which half-wave
    const int l = lane & 15;            // M/N position

    // Per-lane Gaussian weight fragments; identical for horizontal-pass B
    // (Wh[j][c]=g[j-c], c=l) and vertical-pass A (Wv[r][j]=g[j-r], r=l).
    v2f wf[7];
#pragma unroll
    for (int k = 0; k < 7; ++k) {
        const int j = k * 4 + 2 * half;
        wf[k].x = wgt(G, j - l);
        wf[k].y = wgt(G, j + 1 - l);
    }

    // ---- Horizontal pass: one A-load pair feeds all 5 field accumulators ----
#pragma unroll
    for (int rt = 0; rt < 2; ++rt) {
        v8f acc[5];
#pragma unroll
        for (int f = 0; f < 5; ++f)
            acc[f] = (v8f){0.f, 0.f, 0.f, 0.f, 0.f, 0.f, 0.f, 0.f};
        const int row = rt * 16 + l;    // A-matrix row (M) for this lane
#pragma unroll
        for (int k = 0; k < 7; ++k) {
            const int j = k * 4 + 2 * half;  // A K-pair for this half-wave
            const float x0 = Tx[row * 28 + j];
            const float x1 = Tx[row * 28 + j + 1];
            const float y0 = Ty[row * 28 + j];
            const float y1 = Ty[row * 28 + j + 1];
            const v2f ax  = {x0, x1};
            const v2f ay  = {y0, y1};
            const v2f axx = {x0 * x0, x1 * x1};
            const v2f ayy = {y0 * y0, y1 * y1};
            const v2f axy = {x0 * y0, x1 * y1};
            acc[0] = WMMA_F32X4(ax,  wf[k], acc[0]);
            acc[1] = WMMA_F32X4(ay,  wf[k], acc[1]);
            acc[2] = WMMA_F32X4(axx, wf[k], acc[2]);
            acc[3] = WMMA_F32X4(ayy, wf[k], acc[3]);
            acc[4] = WMMA_F32X4(axy, wf[k], acc[4]);
        }
        // Scatter D fragments into transposed H: Ht[f][col=l][row]
#pragma unroll
        for (int f = 0; f < 5; ++f)
#pragma unroll
            for (int v = 0; v < 8; ++v)
                Ht[f * 544 + l * 34 + (rt * 16 + v + 8 * half)] = acc[f][v];
    }

    // ---- Vertical pass per field: Out = Wv(16x28) x H(28x16) ----
    v8f res[5];
#pragma unroll
    for (int f = 0; f < 5; ++f) {
        v8f o = {0.f, 0.f, 0.f, 0.f, 0.f, 0.f, 0.f, 0.f};
        const float* Hf = Ht + f * 544;
#pragma unroll
        for (int k = 0; k < 7; ++k) {
            const int j = k * 4 + 2 * half;
            v2f b;                       // rows j, j+1 of column l: contiguous
            b.x = Hf[l * 34 + j];
            b.y = Hf[l * 34 + j + 1];
            o = WMMA_F32X4(wf[k], b, o);
        }
        res[f] = o;
    }

    // ---- Pointwise SSIM map + partial sum over this lane's 8 elements ----
    const float C1 = 1e-4f;   // 0.01^2
    const float C2 = 9e-4f;   // 0.03^2
    float sum = 0.0f;
#pragma unroll
    for (int v = 0; v < 8; ++v) {
        const float m1 = res[0][v];
        const float m2 = res[1][v];
        const float s1 = res[2][v] - m1 * m1;
        const float s2 = res[3][v] - m2 * m2;
        const float s12 = res[4][v] - m1 * m2;
        const float num = (2.0f * m1 * m2 + C1) * (2.0f * s12 + C2);
        const float dnm = (m1 * m1 + m2 * m2 + C1) * (s1 + s2 + C2);
        sum += num / dnm;
    }

    // wave32 reduction + one atomic per tile
    for (int off = 16; off > 0; off >>= 1)
        sum += __shfl_down(sum, off, 32);
    if (lane == 0) atomicAdd(accum, sum);
}

__global__ void ssim_finalize_kernel(const float* __restrict__ accum,
                                     float* __restrict__ out) {
    out[0] = 1.0f - accum[0] * (1.0f / 25165824.0f);  // / (32*3*512*512)
}

extern "C" void kernel_launch(void* const* d_in, const int* in_sizes, int n_in,
                              void* d_out, int out_size, void* d_ws,
                              size_t ws_size, hipStream_t stream) {
    const float* den = (const float*)d_in[0];
    const float* cln = (const float*)d_in[1];
    float* accum = (float*)d_ws;

    hipMemsetAsync(accum, 0, sizeof(float), stream);

    const int nTiles = 32 * 3 * 32 * 32;  // 98304 16x16 tiles
    const int blocks = nTiles / 2;        // 2 waves (tiles) per block
    ssim_wmma_kernel<<<blocks, 64, 0, stream>>>(den, cln, accum, nTiles);
    ssim_finalize_kernel<<<1, 1, 0, stream>>>(accum, (float*)d_out);
}